// PSNR_LCS_22660247453703
// MI455X (gfx1250) — compile-verified
//
#include <hip/hip_runtime.h>
#include <math.h>

// ---------------------------------------------------------------------------
// PSNR + Local-Cosine-Similarity loss, fused, for gfx1250 (MI455X).
// Box filter (11x11, zero pad) done as banded-matrix GEMMs on the fp32 WMMA
// path (v_wmma_f32_16x16x4_f32). Wave32 shuffle reductions, deterministic
// two-stage final reduction (no atomics).
// ---------------------------------------------------------------------------

typedef __attribute__((ext_vector_type(2))) float v2f;
typedef __attribute__((ext_vector_type(8))) float v8f;

#define NB    16
#define NC    3
#define HIMG  512
#define WIMG  512
#define HW    (HIMG * WIMG)
#define KWIN  11
#define PAD   5

#define TO    32            // output tile (32x32)
#define THL   (TO + 10)     // halo tile (42x42)
#define ROWS  48            // rows padded to 3 chunks of 16
#define STR   44            // col stride padded (bank-conflict-free, >= 16+27+1)
#define PLANE (ROWS * STR)  // 2112 floats per map

#define BLOCKS_PER_IMG 256  // 16x16 tiles
#define NBLOCKS (NB * BLOCKS_PER_IMG)  // 4096

__device__ __forceinline__ v8f wmma_f32_k4(v2f a, v2f b, v8f c) {
    // D = A(16x4,f32) * B(4x16,f32) + C(16x16,f32)
    return __builtin_amdgcn_wmma_f32_16x16x4_f32(
        /*neg_a=*/false, a, /*neg_b=*/false, b,
        /*c_mod=*/(short)0, c, /*reuse_a=*/false, /*reuse_b=*/false);
}

__global__ __launch_bounds__(256)
void psnr_lcs_tile_kernel(const float* __restrict__ pred,
                          const float* __restrict__ targ,
                          float* __restrict__ ws) {
    __shared__ float sS[NC * PLANE];   // per-pixel sxy/sxx/syy (halo, padded, zero-filled)
    __shared__ float sH[NC * PLANE];   // horizontal 11-tap sums
    __shared__ float wredM[8];
    __shared__ float wredC[8];

    const int tid  = threadIdx.x;
    const int wave = tid >> 5;
    const int lane = tid & 31;
    const int hh   = lane >> 4;   // half-wave select (K striping)
    const int nn   = lane & 15;   // M for A-data / N for B-data / N for D

    const int b   = blockIdx.z;
    const int ty0 = blockIdx.y * TO - PAD;
    const int tx0 = blockIdx.x * TO - PAD;
    const size_t imgBase = (size_t)b * NC * HW;

    // ---- Phase 1: load halo, form channel-reduced products, MSE partial ----
    float mseAcc = 0.0f;
    for (int idx = tid; idx < PLANE; idx += 256) {
        const int r = idx / STR;
        const int c = idx % STR;
        const int gy = ty0 + r;
        const int gx = tx0 + c;
        float sxy = 0.0f, sxx = 0.0f, syy = 0.0f;
        if (r < THL && c < THL && gy >= 0 && gy < HIMG && gx >= 0 && gx < WIMG) {
            const size_t p = imgBase + (size_t)gy * WIMG + gx;
            const float p0 = pred[p];
            const float p1 = pred[p + HW];
            const float p2 = pred[p + 2 * HW];
            const float t0 = targ[p];
            const float t1 = targ[p + HW];
            const float t2 = targ[p + 2 * HW];
            sxy = p0 * t0 + p1 * t1 + p2 * t2;
            sxx = p0 * p0 + p1 * p1 + p2 * p2;
            syy = t0 * t0 + t1 * t1 + t2 * t2;
            if (r >= PAD && r < PAD + TO && c >= PAD && c < PAD + TO) {
                const float d0 = p0 - t0, d1 = p1 - t1, d2 = p2 - t2;
                mseAcc += d0 * d0 + d1 * d1 + d2 * d2;
            }
        }
        sS[0 * PLANE + idx] = sxy;
        sS[1 * PLANE + idx] = sxx;
        sS[2 * PLANE + idx] = syy;
    }
    __syncthreads();

    // ---- Phase 2: horizontal 11-tap sums via WMMA: D = S(16x28) x Band(28x16)
    // jobs: map(3) x rowChunk(3) x colChunk(2) = 18
    for (int job = wave; job < 18; job += 8) {
        const int mp = job / 6;
        const int rem = job % 6;
        const int rc = rem >> 1;   // row chunk 0..2
        const int cc = rem & 1;    // col chunk 0..1
        const float* sp = &sS[mp * PLANE];
        v8f acc = {};
        const int rowA = rc * 16 + nn;           // A: M = lane%16
        #pragma unroll
        for (int kc = 0; kc < 7; ++kc) {
            const int k0 = kc * 4 + 2 * hh;      // K striping: VGPR0->K=2h, VGPR1->K=2h+1
            v2f a;
            a.x = sp[rowA * STR + cc * 16 + k0];
            a.y = sp[rowA * STR + cc * 16 + k0 + 1];
            v2f bb;                               // Band[k][n] = (n <= k <= n+10)
            bb.x = (k0     >= nn && k0     <= nn + 10) ? 1.0f : 0.0f;
            bb.y = (k0 + 1 >= nn && k0 + 1 <= nn + 10) ? 1.0f : 0.0f;
            acc = wmma_f32_k4(a, bb, acc);
        }
        float* hp = &sH[mp * PLANE];
        #pragma unroll
        for (int v = 0; v < 8; ++v) {            // D: VGPR v -> M = v + 8*half
            const int rowD = rc * 16 + v + 8 * hh;
            hp[rowD * STR + cc * 16 + nn] = acc[v];
        }
    }
    __syncthreads();

    // ---- Phase 3: vertical 11-tap sums via WMMA + cosine ----
    // D = Band(16x28) x H(28x16); jobs: 2x2 output subtiles, all 3 maps per job
    float cosAcc = 0.0f;
    for (int job = wave; job < 4; job += 8) {
        const int tr = job >> 1;
        const int tc = job & 1;
        v8f cxy = {}, cxx = {}, cyy = {};
        #pragma unroll
        for (int kc = 0; kc < 7; ++kc) {
            const int k0 = kc * 4 + 2 * hh;
            v2f a;                                // Band[m][k] = (m <= k <= m+10), m = lane%16
            a.x = (k0     >= nn && k0     <= nn + 10) ? 1.0f : 0.0f;
            a.y = (k0 + 1 >= nn && k0 + 1 <= nn + 10) ? 1.0f : 0.0f;
            const int r0 = tr * 16 + k0;
            const int cb = tc * 16 + nn;
            v2f b0, b1, b2;
            b0.x = sH[0 * PLANE + r0 * STR + cb];
            b0.y = sH[0 * PLANE + (r0 + 1) * STR + cb];
            b1.x = sH[1 * PLANE + r0 * STR + cb];
            b1.y = sH[1 * PLANE + (r0 + 1) * STR + cb];
            b2.x = sH[2 * PLANE + r0 * STR + cb];
            b2.y = sH[2 * PLANE + (r0 + 1) * STR + cb];
            cxy = wmma_f32_k4(a, b0, cxy);
            cxx = wmma_f32_k4(a, b1, cxx);
            cyy = wmma_f32_k4(a, b2, cyy);
        }
        #pragma unroll
        for (int v = 0; v < 8; ++v) {
            const float cosv = cxy[v] / (sqrtf(cxx[v]) * sqrtf(cyy[v]) + 1e-6f);
            cosAcc += cosv;
        }
    }

    // ---- Reduction: wave32 shuffle tree, then fixed-order block combine ----
    #pragma unroll
    for (int off = 16; off >= 1; off >>= 1) {
        mseAcc += __shfl_xor(mseAcc, off, 32);
        cosAcc += __shfl_xor(cosAcc, off, 32);
    }
    if (lane == 0) {
        wredM[wave] = mseAcc;
        wredC[wave] = cosAcc;
    }
    __syncthreads();
    if (tid == 0) {
        float m = 0.0f, cs = 0.0f;
        #pragma unroll
        for (int i = 0; i < 8; ++i) { m += wredM[i]; cs += wredC[i]; }
        const int linear = (blockIdx.z * 16 + blockIdx.y) * 16 + blockIdx.x;
        ws[linear]           = cs;   // cosine partial
        ws[NBLOCKS + linear] = m;    // squared-error partial
    }
}

// Deterministic final reduction + loss assembly. Fixed summation order.
__global__ __launch_bounds__(256)
void psnr_lcs_finalize_kernel(const float* __restrict__ ws,
                              float* __restrict__ out) {
    __shared__ float lP[256];
    __shared__ float mP[256];
    __shared__ float lg[16];
    const int t = threadIdx.x;

    // cosine partials: 4096 -> 256
    float s = 0.0f;
    for (int i = 0; i < 16; ++i) s += ws[t * 16 + i];
    lP[t] = s;

    // mse partials: per batch 256 -> 16 segments of 16
    const int bb  = t >> 4;
    const int seg = t & 15;
    float sm = 0.0f;
    for (int i = 0; i < 16; ++i) sm += ws[NBLOCKS + bb * BLOCKS_PER_IMG + seg * 16 + i];
    mP[t] = sm;
    __syncthreads();

    if (t < 16) {
        float tm = 0.0f;
        for (int i = 0; i < 16; ++i) tm += mP[t * 16 + i];
        const float mse = tm / (float)(NC * HW);
        lg[t] = logf(mse + 1e-8f);
    }
    __syncthreads();

    if (t == 0) {
        float lcsSum = 0.0f;
        for (int i = 0; i < 256; ++i) lcsSum += lP[i];
        float lgs = 0.0f;
        for (int i = 0; i < 16; ++i) lgs += lg[i];
        const float scale = 10.0f / logf(10.0f);
        const float psnr_loss = scale * (lgs / (float)NB);
        const float lcs_loss  = 1.0f - lcsSum / (float)((size_t)NB * HW);
        out[0] = psnr_loss + lcs_loss;   // weights are 1.0, 1.0
    }
}

extern "C" void kernel_launch(void* const* d_in, const int* in_sizes, int n_in,
                              void* d_out, int out_size, void* d_ws, size_t ws_size,
                              hipStream_t stream) {
    (void)in_sizes; (void)n_in; (void)out_size; (void)ws_size;
    const float* pred = (const float*)d_in[0];
    const float* targ = (const float*)d_in[1];
    float* ws  = (float*)d_ws;
    float* out = (float*)d_out;

    dim3 grid(WIMG / TO, HIMG / TO, NB);   // 16 x 16 x 16 = 4096 blocks
    psnr_lcs_tile_kernel<<<grid, 256, 0, stream>>>(pred, targ, ws);
    psnr_lcs_finalize_kernel<<<1, 256, 0, stream>>>(ws, out);
}